// GNNPolicy_31473520345936
// MI455X (gfx1250) — compile-verified
//
#include <hip/hip_runtime.h>
#include <hip/hip_bf16.h>

#define N_NODES 50000
#define N_EDGES 1600000
#define FDIM    128

typedef __attribute__((ext_vector_type(2))) float v2f;
typedef __attribute__((ext_vector_type(8))) float v8f;

// ---------------- utility ----------------
__global__ void zero_f32_kernel(float* __restrict__ p, long n) {
    long i = (long)blockIdx.x * blockDim.x + threadIdx.x;
    long stride = (long)gridDim.x * blockDim.x;
    for (; i < n; i += stride) p[i] = 0.0f;
}

// ---------------- degree ----------------
__global__ void deg_accum_kernel(const int* __restrict__ dst, float* __restrict__ degbuf, int e_count) {
    int e = blockIdx.x * blockDim.x + threadIdx.x;
    if (e < e_count) atomicAdd(&degbuf[dst[e]], 1.0f);
}

__global__ void finish_dinv_kernel(float* __restrict__ degbuf, int n) {
    int i = blockIdx.x * blockDim.x + threadIdx.x;
    if (i < n) degbuf[i] = rsqrtf(degbuf[i] + 1.0f);
}

// ---------------- edge scatter: agg[dst] += x[src] * dinv[src] ----------------
// 2 edges per 256-thread block; each edge handled by 128 contiguous threads
// -> each edge touches exactly one contiguous 512B row per array.
__global__ __launch_bounds__(256)
void gcn_scatter_kernel(const float* __restrict__ x, const int* __restrict__ src,
                        const int* __restrict__ dst, const float* __restrict__ dinv,
                        float* __restrict__ agg) {
    int e = blockIdx.x * 2 + (threadIdx.x >> 7);
    if (e >= N_EDGES) return;
    int f = threadIdx.x & 127;
    int s = src[e];
    int d = dst[e];
    float w = dinv[s];
    atomicAdd(&agg[(size_t)d * FDIM + f], x[(size_t)s * FDIM + f] * w);
}

// ---------------- fused combine + GEMM + ReLU (WMMA f32 16x16x4) ----------------
// out[n,:] = relu( ((agg[n,:] + x[n,:]*dinv[n]) * dinv[n]) @ W )
// Block: 16 rows of the node dim, 256 threads = 8 wave32; wave w computes
// output columns [16w, 16w+16). K = 128 -> 32 x v_wmma_f32_16x16x4_f32.
// `out` may alias `agg`: each block fully consumes its agg rows into LDS
// (and __syncthreads) before any store, and blocks own disjoint row sets.
__global__ __launch_bounds__(256)
void gcn_combine_gemm_relu_kernel(const float* __restrict__ agg,
                                  const float* __restrict__ x,
                                  const float* __restrict__ dinv,
                                  const float* __restrict__ W,   // [128 x 128] row-major (k, j)
                                  float* out) {
    __shared__ float As[16][132];   // pad 128->132 to spread LDS banks
    const int tid = threadIdx.x;
    const int rowBase = blockIdx.x * 16;

    // Cooperative load+combine of the 16x128 A tile (2048 elems / 256 thr = 8 each).
#pragma unroll
    for (int i = 0; i < 8; ++i) {
        int idx = tid + 256 * i;
        int r = idx >> 7;
        int c = idx & 127;
        int n = rowBase + r;
        float di = dinv[n];
        As[r][c] = (agg[(size_t)n * FDIM + c] + x[(size_t)n * FDIM + c] * di) * di;
    }
    __syncthreads();

    const int wave = tid >> 5;            // 0..7 -> output col tile
    const int lane = tid & 31;
    const int m    = lane & 15;           // A row / B column within tile
    const int kh   = (lane >> 4) << 1;    // K sub-offset: lanes 0-15 -> K+0/K+1, 16-31 -> K+2/K+3
    const int col  = (wave << 4) + m;     // global output column

    v8f acc = {};
#pragma unroll
    for (int kk = 0; kk < FDIM; kk += 4) {
        v2f a, b;
        // A 16x4 f32 layout: VGPR0 = K (kh), VGPR1 = K+1; M striped over lanes%16
        a.x = As[m][kk + kh];
        a.y = As[m][kk + kh + 1];
        // B 4x16 f32 layout: row K striped across lanes within a VGPR
        b.x = W[(size_t)(kk + kh) * FDIM + col];
        b.y = W[(size_t)(kk + kh + 1) * FDIM + col];
        acc = __builtin_amdgcn_wmma_f32_16x16x4_f32(
            /*neg_a=*/false, a, /*neg_b=*/false, b,
            /*c_mod=*/(short)0, acc, /*reuse_a=*/false, /*reuse_b=*/false);
    }

    // C/D layout: VGPR r holds M = r + 8*(lane/16), N = lane%16
    const int mBase = rowBase + ((lane >> 4) << 3);
#pragma unroll
    for (int r = 0; r < 8; ++r) {
        float v = acc[r];
        v = v > 0.0f ? v : 0.0f;
        out[(size_t)(mBase + r) * FDIM + col] = v;
    }
}

// ---------------- column sum (for mean) ----------------
__global__ __launch_bounds__(128)
void col_sum_kernel(const float* __restrict__ h, float* __restrict__ gbuf, int nrows) {
    int f = threadIdx.x;
    int r0 = blockIdx.x * 128;
    int rend = r0 + 128;
    if (rend > nrows) rend = nrows;
    float s = 0.0f;
    for (int r = r0; r < rend; ++r) s += h[(size_t)r * FDIM + f];
    atomicAdd(&gbuf[f], s);
}

// ---------------- gdot = (g/N) . wg + b ----------------
__global__ __launch_bounds__(128)
void gdot_kernel(const float* __restrict__ gbuf, const float* __restrict__ wg,
                 const float* __restrict__ b_edge, float* __restrict__ gdot) {
    __shared__ float red[128];
    int t = threadIdx.x;
    red[t] = (gbuf[t] * (1.0f / (float)N_NODES)) * wg[t];
    __syncthreads();
    for (int s = 64; s > 0; s >>= 1) {
        if (t < s) red[t] += red[t + s];
        __syncthreads();
    }
    if (t == 0) *gdot = red[0] + b_edge[0];
}

// ---------------- per-node scores: s_src[n] = h[n].ws, s_dst[n] = h[n].wd ----------------
__global__ __launch_bounds__(128)
void node_scores_kernel(const float* __restrict__ h, const float* __restrict__ wsv,
                        const float* __restrict__ wdv, float* __restrict__ ssrc,
                        float* __restrict__ sdst) {
    __shared__ float r1[128];
    __shared__ float r2[128];
    int n = blockIdx.x;
    int t = threadIdx.x;
    float v = h[(size_t)n * FDIM + t];
    r1[t] = v * wsv[t];
    r2[t] = v * wdv[t];
    __syncthreads();
    for (int s = 64; s > 0; s >>= 1) {
        if (t < s) { r1[t] += r1[t + s]; r2[t] += r2[t + s]; }
        __syncthreads();
    }
    if (t == 0) { ssrc[n] = r1[0]; sdst[n] = r2[0]; }
}

// ---------------- edge logits ----------------
__global__ __launch_bounds__(256)
void edge_logits_kernel(const int* __restrict__ src, const int* __restrict__ dst,
                        const float* __restrict__ ssrc, const float* __restrict__ sdst,
                        const float* __restrict__ gdot, float* __restrict__ out, int e_count) {
    int e = blockIdx.x * blockDim.x + threadIdx.x;
    if (e < e_count) out[e] = *gdot + ssrc[src[e]] + sdst[dst[e]];
}

extern "C" void kernel_launch(void* const* d_in, const int* in_sizes, int n_in,
                              void* d_out, int out_size, void* d_ws, size_t ws_size,
                              hipStream_t stream) {
    const float* x   = (const float*)d_in[0];   // [N,128]
    const float* W1  = (const float*)d_in[1];   // [128,128]
    const float* W2  = (const float*)d_in[2];   // [128,128]
    const float* wE  = (const float*)d_in[3];   // [384] = wg|ws|wd
    const float* bE  = (const float*)d_in[4];   // scalar
    const int*   ei  = (const int*)d_in[5];     // [2,E]
    (void)in_sizes; (void)n_in; (void)out_size; (void)ws_size;

    const int* src = ei;
    const int* dst = ei + N_EDGES;

    // workspace layout (floats)
    float* ws   = (float*)d_ws;
    float* dinv = ws;                                 // N   (also deg accumulator)
    float* agg  = dinv + N_NODES;                     // N*128 (also layer-2 output h2)
    float* h1   = agg + (size_t)N_NODES * FDIM;       // N*128
    float* gbuf = h1 + (size_t)N_NODES * FDIM;        // 128
    float* gdot = gbuf + FDIM;                        // 1
    float* ssrc = gdot + 1;                           // N
    float* sdst = ssrc + N_NODES;                     // N
    float* h2   = agg;                                // alias (safe, see kernel comment)

    const int eBlocks256 = (N_EDGES + 255) / 256;
    const int nBlocks256 = (N_NODES + 255) / 256;

    // deg -> dinv
    zero_f32_kernel<<<2048, 256, 0, stream>>>(dinv, (long)N_NODES * (1 + FDIM)); // dinv + agg contiguous
    zero_f32_kernel<<<1, 256, 0, stream>>>(gbuf, FDIM + 1);
    deg_accum_kernel<<<eBlocks256, 256, 0, stream>>>(dst, dinv, N_EDGES);
    finish_dinv_kernel<<<nBlocks256, 256, 0, stream>>>(dinv, N_NODES);

    // layer 1
    gcn_scatter_kernel<<<(N_EDGES + 1) / 2, 256, 0, stream>>>(x, src, dst, dinv, agg);
    gcn_combine_gemm_relu_kernel<<<N_NODES / 16, 256, 0, stream>>>(agg, x, dinv, W1, h1);

    // layer 2 (reuse agg; output aliases agg)
    zero_f32_kernel<<<2048, 256, 0, stream>>>(agg, (long)N_NODES * FDIM);
    gcn_scatter_kernel<<<(N_EDGES + 1) / 2, 256, 0, stream>>>(h1, src, dst, dinv, agg);
    gcn_combine_gemm_relu_kernel<<<N_NODES / 16, 256, 0, stream>>>(agg, h1, dinv, W2, h2);

    // head
    col_sum_kernel<<<(N_NODES + 127) / 128, 128, 0, stream>>>(h2, gbuf, N_NODES);
    gdot_kernel<<<1, 128, 0, stream>>>(gbuf, wE /*wg*/, bE, gdot);
    node_scores_kernel<<<N_NODES, 128, 0, stream>>>(h2, wE + FDIM /*ws*/, wE + 2 * FDIM /*wd*/, ssrc, sdst);
    edge_logits_kernel<<<eBlocks256, 256, 0, stream>>>(src, dst, ssrc, sdst, gdot, (float*)d_out, N_EDGES);
}